// ModelConTT_46016279609475
// MI455X (gfx1250) — compile-verified
//
#include <hip/hip_runtime.h>

// CDNA5 / gfx1250 TT-interpolation evaluator.
//   y[b] = v0(b)^T * M1(b) * M2(b) * v3(b), slices linearly interpolated.
// Strategy: stage all TT cores (272KB) into LDS per workgroup; one point per
// wave32; build interpolated M1/M2 slices directly in WMMA A/B register
// layouts (core2 stored transposed in LDS so both operands use contiguous
// 8-byte LDS loads); P = M1*M2 via 4x v_wmma_f32_16x16x4_f32; epilogue
// contracts P with v0/v3 per-lane + wave32 xor-shuffle reduction.

typedef float v2f __attribute__((ext_vector_type(2)));
typedef float v8f __attribute__((ext_vector_type(8)));

#define NDIM 4
#define NS   128
#define RK   16

// LDS layout (in floats)
constexpr int S1_OFF = 0;                         // core1 [rp][n][rn]   : 16*128*16
constexpr int S2_OFF = S1_OFF + RK * NS * RK;     // core2T [rn][n][rp]  : 16*128*16
constexpr int S0_OFF = S2_OFF + RK * NS * RK;     // core0 [n][r]        : 128*16
constexpr int S3_OFF = S0_OFF + NS * RK;          // core3 [r][n]        : 16*128
constexpr int SMEM_FLOATS = S3_OFF + RK * NS;     // total = 69632 floats = 272KB

__global__ void __launch_bounds__(1024)
tt_eval_kernel(const float* __restrict__ x,
               const float* __restrict__ c0,
               const float* __restrict__ c1,
               const float* __restrict__ c2,
               const float* __restrict__ c3,
               float* __restrict__ out, int B)
{
    extern __shared__ float smem[];
    float* s1 = smem + S1_OFF;
    float* s2 = smem + S2_OFF;
    float* s0 = smem + S0_OFF;
    float* s3 = smem + S3_OFF;

    const int tid  = threadIdx.x;
    const int nthr = blockDim.x;

    // ---- stage cores into LDS (core2 transposed for contiguous B-operand) ----
    for (int i = tid; i < RK * NS * RK; i += nthr) s1[i] = c1[i];
    for (int i = tid; i < RK * NS * RK; i += nthr) {
        const int rn = i & (RK - 1);
        const int n  = (i >> 4) & (NS - 1);
        const int rp = i >> 11;
        s2[(rn * NS + n) * RK + rp] = c2[i];
    }
    for (int i = tid; i < NS * RK; i += nthr) s0[i] = c0[i];
    for (int i = tid; i < RK * NS; i += nthr) s3[i] = c3[i];
    __syncthreads();

    const int lane = tid & 31;
    const int half = lane >> 4;   // 0: lanes 0-15, 1: lanes 16-31
    const int l16  = lane & 15;

    const int wave   = blockIdx.x * (nthr >> 5) + (tid >> 5);
    const int nwaves = gridDim.x * (nthr >> 5);

    for (int p = wave; p < B; p += nwaves) {
        // ---- per-dim gather indices + interpolation weights (wave-uniform) ----
        int   lo[NDIM], hi[NDIM];
        float w[NDIM];
#pragma unroll
        for (int d = 0; d < NDIM; ++d) {
            float xc = (x[p * NDIM + d] + 1.0f) * (0.5f * (float)(NS - 1));
            xc = fminf(fmaxf(xc, 0.0f), (float)(NS - 1));
            const float fl = floorf(xc);
            lo[d] = (int)fl;
            hi[d] = (int)ceilf(xc);
            w[d]  = xc - fl;
        }

        // ---- P = M1i * M2i via 4 chained f32 WMMAs (K = 16 in steps of 4) ----
        v8f acc = {};
#pragma unroll
        for (int kk = 0; kk < 4; ++kk) {
            const int k0 = kk * 4 + 2 * half;  // even -> 8B-aligned LDS loads

            // A operand: M1i[row=l16][k0 .. k0+1]
            const v2f alo = *(const v2f*)(s1 + (l16 * NS + lo[1]) * RK + k0);
            const v2f ahi = *(const v2f*)(s1 + (l16 * NS + hi[1]) * RK + k0);
            const v2f Av  = alo + (ahi - alo) * w[1];

            // B operand: M2i[k0 .. k0+1][col=l16]  (core2 stored transposed)
            const v2f blo = *(const v2f*)(s2 + (l16 * NS + lo[2]) * RK + k0);
            const v2f bhi = *(const v2f*)(s2 + (l16 * NS + hi[2]) * RK + k0);
            const v2f Bv  = blo + (bhi - blo) * w[2];

            acc = __builtin_amdgcn_wmma_f32_16x16x4_f32(
                /*neg_a=*/false, Av, /*neg_b=*/false, Bv,
                /*c_mod=*/(short)0, acc, /*reuse_a=*/false, /*reuse_b=*/false);
        }

        // ---- epilogue: y = sum_{m,n} v0i[m] * P[m][n] * v3i[n] ----
        // C/D layout: lane holds P[v + 8*half][l16] in acc[v], v = 0..7
        const float t3lo = s3[l16 * NS + lo[3]];
        const float t3hi = s3[l16 * NS + hi[3]];
        const float v3n  = t3lo + (t3hi - t3lo) * w[3];

        float sum = 0.0f;
#pragma unroll
        for (int v = 0; v < 8; ++v) {
            const int m   = v + 8 * half;
            const float a = s0[lo[0] * RK + m];
            const float b = s0[hi[0] * RK + m];
            const float v0m = a + (b - a) * w[0];
            sum += v0m * acc[v];
        }
        float contrib = sum * v3n;

        // wave32 reduction
#pragma unroll
        for (int off = 16; off > 0; off >>= 1)
            contrib += __shfl_xor(contrib, off, 32);

        if (lane == 0) out[p] = contrib;
    }
}

extern "C" void kernel_launch(void* const* d_in, const int* in_sizes, int n_in,
                              void* d_out, int out_size, void* d_ws, size_t ws_size,
                              hipStream_t stream) {
    const float* x  = (const float*)d_in[0];
    const float* c0 = (const float*)d_in[1];
    const float* c1 = (const float*)d_in[2];
    const float* c2 = (const float*)d_in[3];
    const float* c3 = (const float*)d_in[4];
    float* out = (float*)d_out;

    const int B = in_sizes[0] / NDIM;

    const size_t shmem = (size_t)SMEM_FLOATS * sizeof(float);  // 272 KB (<= 320 KB/WGP)
    (void)hipFuncSetAttribute((const void*)tt_eval_kernel,
                              hipFuncAttributeMaxDynamicSharedMemorySize,
                              (int)shmem);

    // 1024 threads = 32 waves/WG, 1 WG per WGP (LDS-bound); grid-stride over points.
    const dim3 block(1024);
    const dim3 grid(512);
    hipLaunchKernelGGL(tt_eval_kernel, grid, block, shmem, stream,
                       x, c0, c1, c2, c3, out, B);
}